// CharRNN_25297357373666
// MI455X (gfx1250) — compile-verified
//
#include <hip/hip_runtime.h>
#include <hip/hip_bf16.h>

typedef __attribute__((ext_vector_type(2))) float v2f;
typedef __attribute__((ext_vector_type(8))) float v8f;
typedef __attribute__((ext_vector_type(4))) int   v4i;

#define B_   1024
#define S_   256
#define V_   256
#define H_   10
#define G_   40    // 4*H
#define L_   15
#define KPAD 12    // H padded to multiple of 4 (K dim for 16x16x4 f32 WMMA)
#define NPAD 48    // 4H padded to multiple of 16 (3 N-tiles)

__device__ __forceinline__ float fast_sigmoid(float x) {
    // rcp-based: sigma(x) = rcp(1 + exp(-x))   (v_exp_f32 + v_rcp_f32, no div expansion)
    return __builtin_amdgcn_rcpf(1.0f + __expf(-x));
}
__device__ __forceinline__ float fast_tanh(float x) {
    // tanh(x) = 1 - 2*rcp(exp(2x) + 1)
    float r = __builtin_amdgcn_rcpf(__expf(2.0f * x) + 1.0f);
    return fmaf(-2.0f, r, 1.0f);
}

__global__ __launch_bounds__(32) void charrnn_lstm_wmma(
    const int*   __restrict__ x,    // [B,S]
    const float* __restrict__ Wx,   // [V,4H]
    const float* __restrict__ Wh,   // [H,4H]
    const float* __restrict__ bias, // [4H]
    const float* __restrict__ Wd,   // [H,L]
    const float* __restrict__ bd,   // [L]
    const float* __restrict__ drop_rate, // scalar
    float*       __restrict__ out)  // [B,L]
{
    __shared__ float sWx[V_ * NPAD]; // 48 KB gather table, cols 40..47 zero-padded
    __shared__ int   sx[S_ * 16];    // 16 KB PRE-SCALED row offsets: token * NPAD
    __shared__ float sh[16 * KPAD];  // h state, row-major padded (A-matrix source)
    __shared__ float sg[16 * NPAD];  // gate staging (D-layout dump)

    const int lane  = threadIdx.x;       // wave32
    const int row0  = blockIdx.x * 16;   // 16 batch rows per wave
    const int lhalf = lane >> 4;         // 0: lanes 0-15, 1: lanes 16-31
    const int lmod  = lane & 15;

    // ---- cooperative staging (once) ----
    for (int i = lane; i < V_ * NPAD; i += 32) {
        int r = i / NPAD, n = i % NPAD;
        sWx[i] = (n < G_) ? Wx[r * G_ + n] : 0.0f;
    }
    for (int i = lane; i < 16 * S_; i += 32) {
        int m = i / S_, t = i % S_;                       // coalesced global read
        sx[t * 16 + m] = x[(row0 + m) * S_ + t] * NPAD;   // transposed + pre-scaled
    }
    for (int i = lane; i < 16 * KPAD; i += 32) sh[i] = 0.0f;  // h0 = 0 (incl. pad)
    __syncthreads();

    // ---- Wh in WMMA B-matrix layout, kept in registers (constant over scan) ----
    // B 4x16 f32: vgpr j, lanes 0-15 -> B[K=j][n], lanes 16-31 -> B[K=j+2][n]
    v2f bm[3][3];
    #pragma unroll
    for (int kt = 0; kt < 3; ++kt) {
        #pragma unroll
        for (int nt = 0; nt < 3; ++nt) {
            int k0 = 4 * kt + 2 * lhalf;
            int n  = 16 * nt + lmod;
            float e0 = (k0     < H_ && n < G_) ? Wh[k0 * G_ + n]       : 0.0f;
            float e1 = (k0 + 1 < H_ && n < G_) ? Wh[(k0 + 1) * G_ + n] : 0.0f;
            v2f tv; tv.x = e0; tv.y = e1;
            bm[kt][nt] = tv;
        }
    }

    // per-lane column constants (zero-padded bias -> branch-free C build)
    float biasv[3];
    int   ncol[3];
    #pragma unroll
    for (int nt = 0; nt < 3; ++nt) {
        int n = 16 * nt + lmod;
        ncol[nt]  = n;
        biasv[nt] = (n < G_) ? bias[n] : 0.0f;
    }

    // hoisted gate-phase (m,j) strips: p = lane + 32u, u = 0..4 (160 = 5*32 exact)
    int goff[5], hoff[5];
    float creg[5];                       // cell state lives in registers (lane-private)
    #pragma unroll
    for (int u = 0; u < 5; ++u) {
        int p = lane + 32 * u;
        int m = p / H_, j = p - m * H_;
        goff[u] = m * NPAD + j;   // gates base
        hoff[u] = m * KPAD + j;   // h write
        creg[u] = 0.0f;           // c0 = 0
    }

    const float inv_keep = 1.0f / (1.0f - drop_rate[0]);
    const int xrow = 8 * lhalf;   // this half-wave's first D-row

    // ---- C-tile builder: xp(t) = Wx[x[m,t], n] + b[n] in D layout ----
    auto build_C = [&](int t, v8f d[3]) {
        v4i xa = *(const v4i*)&sx[t * 16 + xrow];      // pre-scaled row offsets
        v4i xb = *(const v4i*)&sx[t * 16 + xrow + 4];
        int xi[8] = { xa.x, xa.y, xa.z, xa.w, xb.x, xb.y, xb.z, xb.w };
        #pragma unroll
        for (int nt = 0; nt < 3; ++nt) {
            #pragma unroll
            for (int v = 0; v < 8; ++v)
                d[nt][v] = sWx[xi[v] + ncol[nt]] + biasv[nt];
        }
    };

    // ---- sequential scan over S timesteps (software-pipelined C build) ----
    v8f d[3];
    build_C(0, d);

    #pragma unroll 1
    for (int t = 0; t < S_; ++t) {
        // A tiles from h state: lane<16 -> (h[m][4kt],h[m][4kt+1]); lane>=16 -> K+2
        v2f a[3];
        #pragma unroll
        for (int kt = 0; kt < 3; ++kt)
            a[kt] = *(const v2f*)&sh[lmod * KPAD + 4 * kt + 2 * lhalf];

        // gates = xp + h @ Wh : 9 chained f32 WMMAs (exact f32, matches reference)
        #pragma unroll
        for (int nt = 0; nt < 3; ++nt) {
            #pragma unroll
            for (int kt = 0; kt < 3; ++kt) {
                d[nt] = __builtin_amdgcn_wmma_f32_16x16x4_f32(
                    false, a[kt], false, bm[kt][nt],
                    (short)0, d[nt], false, false);
            }
        }

        // dump D tiles to LDS for cross-lane gate combination
        #pragma unroll
        for (int nt = 0; nt < 3; ++nt) {
            #pragma unroll
            for (int v = 0; v < 8; ++v)
                sg[(v + xrow) * NPAD + 16 * nt + lmod] = d[nt][v];
        }

        // prefetch next step's C tiles (independent work to hide DS waits)
        if (t + 1 < S_) build_C(t + 1, d);

        __syncthreads();  // 1-wave WG: compiler fence; HW LDS ops are in-order

        // gate math, fully unrolled strips; Keras gate order i,f,g,o
        #pragma unroll
        for (int u = 0; u < 5; ++u) {
            float gi = sg[goff[u] + 0 * H_];
            float gf = sg[goff[u] + 1 * H_];
            float gg = sg[goff[u] + 2 * H_];
            float go = sg[goff[u] + 3 * H_];
            float i_ = fast_sigmoid(gi);
            float f_ = fast_sigmoid(gf);
            float g_ = fast_tanh(gg);
            float o_ = fast_sigmoid(go);
            float c_ = fmaf(f_, creg[u], i_ * g_);
            creg[u] = c_;
            sh[hoff[u]] = o_ * fast_tanh(c_);   // pad cols stay 0
        }
        __syncthreads();
    }

    // ---- final dense: logits = (h / keep) @ Wd + bd  (16x15, scalar is fine) ----
    for (int p = lane; p < 16 * L_; p += 32) {
        int m = p / L_, l = p % L_;
        float acc = bd[l];
        #pragma unroll
        for (int j = 0; j < H_; ++j)
            acc += (sh[m * KPAD + j] * inv_keep) * Wd[j * L_ + l];
        out[(row0 + m) * L_ + l] = acc;
    }
}

extern "C" void kernel_launch(void* const* d_in, const int* in_sizes, int n_in,
                              void* d_out, int out_size, void* d_ws, size_t ws_size,
                              hipStream_t stream) {
    (void)in_sizes; (void)n_in; (void)out_size; (void)d_ws; (void)ws_size;
    const int*   x         = (const int*)  d_in[0];
    const float* Wx        = (const float*)d_in[1];
    const float* Wh        = (const float*)d_in[2];
    const float* b         = (const float*)d_in[3];
    const float* Wd        = (const float*)d_in[4];
    const float* bd        = (const float*)d_in[5];
    const float* drop_rate = (const float*)d_in[6];
    float* out = (float*)d_out;

    dim3 grid(B_ / 16);  // 64 single-wave workgroups -> spread over WGPs
    dim3 block(32);
    hipLaunchKernelGGL(charrnn_lstm_wmma, grid, block, 0, stream,
                       x, Wx, Wh, b, Wd, bd, drop_rate, out);
}